// FraudDetectionHybrid_65481071410117
// MI455X (gfx1250) — compile-verified
//
#include <hip/hip_runtime.h>
#include <math.h>

// CDNA5 / gfx1250: wave32, WMMA f32 16x16x4, hardware v_tanh_f32.
typedef float v2f __attribute__((ext_vector_type(2)));
typedef float v8f __attribute__((ext_vector_type(8)));

#define NBLK 2048
#define NTHR 256
#define EPSD 1e-5

// Single TRANS-op tanh on gfx1250; branch-free fallback otherwise.
__device__ __forceinline__ float fast_tanh(float x) {
#if __has_builtin(__builtin_amdgcn_tanhf)
    return __builtin_amdgcn_tanhf(x);                 // v_tanh_f32
#else
    // tanh(x) = sign(x) * (1 - 2/(exp2(2*log2e*|x|) + 1)), branch-free
    const float ax = __builtin_fabsf(x);
    const float e  = __builtin_amdgcn_exp2f(ax * 2.8853900817779268f);
    const float t  = 1.0f - 2.0f * __builtin_amdgcn_rcpf(e + 1.0f);
    return __builtin_copysignf(t, x);
#endif
}

// One fused pass: recompute layers 0..NL-1 from x using folded params
// (u = W' h + b', y = tanh(u)) entirely in registers via chained WMMA.
//   A (16x4) = W'  : lane0 row0, lane1 row1, rest zero; K=2,3 zero.
//   B (4x16) = 16 rows of h : VGPR0 lanes0-15 = h0, VGPR1 lanes0-15 = h1,
//              lanes16-31 (K=2,3) zero.
//   C        = bias broadcast on rows M=0,1 only.
//   D layout == next B layout (lanes16-31 stay exactly 0 since tanh(0)=0),
//   so layers chain with no cross-lane shuffles.
// Two independent 16-row chains per wave-iteration for ILP across the
// serial wmma->tanh->wmma dependency. All row indexing is 32-bit.
template<int NL, bool FINAL>
__global__ __launch_bounds__(NTHR)
void chain_pass(const float* __restrict__ x,
                const float* __restrict__ params,
                double* __restrict__ partials,
                float* __restrict__ out,
                int nrows)
{
    const int tid  = threadIdx.x;
    const int lane = tid & 31;

    // Per-layer A-matrix and C-matrix lane values (built once).
    float a0[NL], a1[NL], c0[NL], c1[NL];
#pragma unroll
    for (int j = 0; j < NL; ++j) {
        const float* p = params + 6 * j;            // [w00,w01,w10,w11,b0,b1]
        a0[j] = (lane == 0) ? p[0] : ((lane == 1) ? p[2] : 0.0f);  // K=0 col
        a1[j] = (lane == 0) ? p[1] : ((lane == 1) ? p[3] : 0.0f);  // K=1 col
        c0[j] = (lane < 16) ? p[4] : 0.0f;   // bias row M=0 (M=8 half stays 0)
        c1[j] = (lane < 16) ? p[5] : 0.0f;   // bias row M=1 (M=9 half stays 0)
    }
    float wf0 = 0.0f, wf1 = 0.0f, bfv = 0.0f;
    if constexpr (FINAL) { wf0 = params[102]; wf1 = params[103]; bfv = params[104]; }

    const int wave    = (int)(((unsigned)blockIdx.x * NTHR + (unsigned)tid) >> 5);
    const int nwaves  = (int)(((unsigned)gridDim.x * NTHR) >> 5);
    const int ngroups = (nrows + 15) >> 4;          // 16 rows per chain
    const int npair   = (ngroups + 1) >> 1;         // 2 chains per iteration
    const float2* __restrict__ hp = (const float2*)x;

    double s0 = 0.0, s1 = 0.0, q0 = 0.0, q1 = 0.0;

    for (int i = wave; i < npair; i += nwaves) {
        const int r0 = (i << 5) + lane;             // rows fit in int32
        const int r1 = r0 + 16;
        const bool ok0 = (lane < 16) && (r0 < nrows);
        const bool ok1 = (lane < 16) && (r1 < nrows);
        float u0 = 0.0f, u1 = 0.0f;                 // chain 0 activations
        float w0 = 0.0f, w1 = 0.0f;                 // chain 1 activations
        if (ok0) { float2 h = hp[r0]; u0 = h.x; u1 = h.y; }   // coalesced b64
        if (ok1) { float2 h = hp[r1]; w0 = h.x; w1 = h.y; }

#pragma unroll
        for (int j = 0; j < NL; ++j) {
            v2f A; A[0] = a0[j]; A[1] = a1[j];
            v8f C = {0.f, 0.f, 0.f, 0.f, 0.f, 0.f, 0.f, 0.f};
            C[0] = c0[j]; C[1] = c1[j];
            v2f B0; B0[0] = u0; B0[1] = u1;
            v2f B1; B1[0] = w0; B1[1] = w1;
            // v_wmma_f32_16x16x4_f32 : D = A*B + C
            v8f D0 = __builtin_amdgcn_wmma_f32_16x16x4_f32(
                         false, A, false, B0, (short)0, C, false, false);
            v8f D1 = __builtin_amdgcn_wmma_f32_16x16x4_f32(
                         false, A, false, B1, (short)0, C, false, false);
            u0 = fast_tanh(D0[0]); u1 = fast_tanh(D0[1]);
            w0 = fast_tanh(D1[0]); w1 = fast_tanh(D1[1]);
        }

        if constexpr (FINAL) {
            if (ok0) out[r0] = u0 * wf0 + u1 * wf1 + bfv;
            if (ok1) out[r1] = w0 * wf0 + w1 * wf1 + bfv;
        } else {
            const double m0 = ok0 ? 1.0 : 0.0;      // branch-free tail mask
            const double m1 = ok1 ? 1.0 : 0.0;
            s0 += m0 * (double)u0 + m1 * (double)w0;
            s1 += m0 * (double)u1 + m1 * (double)w1;
            q0 += m0 * (double)u0 * (double)u0 + m1 * (double)w0 * (double)w0;
            q1 += m0 * (double)u1 * (double)u1 + m1 * (double)w1 * (double)w1;
        }
    }

    if constexpr (!FINAL) {
        __shared__ double r0s[NTHR], r1s[NTHR], r2s[NTHR], r3s[NTHR];
        r0s[tid] = s0; r1s[tid] = s1; r2s[tid] = q0; r3s[tid] = q1;
        __syncthreads();
#pragma unroll
        for (int off = NTHR / 2; off > 0; off >>= 1) {
            if (tid < off) {
                r0s[tid] += r0s[tid + off]; r1s[tid] += r1s[tid + off];
                r2s[tid] += r2s[tid + off]; r3s[tid] += r3s[tid + off];
            }
            __syncthreads();
        }
        if (tid == 0) {
            double* pb = partials + (size_t)blockIdx.x * 4;
            pb[0] = r0s[0]; pb[1] = r1s[0]; pb[2] = r2s[0]; pb[3] = r3s[0];
        }
    }
}

// Deterministic cross-block reduction of layer-k stats, then fold
// BN(gamma,beta) + ScaleShift(scale,shift) into the next linear layer
// (or into the final Linear(2,1) when k == 16).
__global__ __launch_bounds__(NTHR)
void fold_kernel(const double* __restrict__ partials, int nparts,
                 const float* __restrict__ W,  const float* __restrict__ bl,
                 const float* __restrict__ gm, const float* __restrict__ bt,
                 const float* __restrict__ sc, const float* __restrict__ sh,
                 const float* __restrict__ Wf, const float* __restrict__ bf,
                 float* __restrict__ params, int k, double invN)
{
    __shared__ double r0[NTHR], r1[NTHR], r2[NTHR], r3[NTHR];
    const int tid = threadIdx.x;
    double s0 = 0.0, s1 = 0.0, q0 = 0.0, q1 = 0.0;
    for (int i = tid; i < nparts; i += NTHR) {
        const double* pb = partials + (size_t)i * 4;
        s0 += pb[0]; s1 += pb[1]; q0 += pb[2]; q1 += pb[3];
    }
    r0[tid] = s0; r1[tid] = s1; r2[tid] = q0; r3[tid] = q1;
    __syncthreads();
    for (int off = NTHR / 2; off > 0; off >>= 1) {
        if (tid < off) {
            r0[tid] += r0[tid + off]; r1[tid] += r1[tid + off];
            r2[tid] += r2[tid + off]; r3[tid] += r3[tid + off];
        }
        __syncthreads();
    }
    if (tid == 0) {
        const double mu0 = r0[0] * invN, mu1 = r1[0] * invN;
        double v0 = r2[0] * invN - mu0 * mu0; if (v0 < 0.0) v0 = 0.0;
        double v1 = r3[0] * invN - mu1 * mu1; if (v1 < 0.0) v1 = 0.0;
        const double i0 = 1.0 / sqrt(v0 + EPSD);
        const double i1 = 1.0 / sqrt(v1 + EPSD);
        const double g0 = (double)gm[2*k],   g1 = (double)gm[2*k+1];
        const double a0 = g0 * i0 * (double)sc[2*k];
        const double a1 = g1 * i1 * (double)sc[2*k+1];
        const double c0 = ((double)bt[2*k]   - mu0 * i0 * g0) * (double)sc[2*k]   + (double)sh[2*k];
        const double c1 = ((double)bt[2*k+1] - mu1 * i1 * g1) * (double)sc[2*k+1] + (double)sh[2*k+1];
        if (k < 16) {
            const float* Wn = W  + (size_t)(k + 1) * 4;   // next layer (2x2)
            const float* bn = bl + (size_t)(k + 1) * 2;
            float* ps = params + 6 * (k + 1);
            ps[0] = (float)((double)Wn[0] * a0);
            ps[1] = (float)((double)Wn[1] * a1);
            ps[2] = (float)((double)Wn[2] * a0);
            ps[3] = (float)((double)Wn[3] * a1);
            ps[4] = (float)((double)bn[0] + (double)Wn[0] * c0 + (double)Wn[1] * c1);
            ps[5] = (float)((double)bn[1] + (double)Wn[2] * c0 + (double)Wn[3] * c1);
        } else {                                          // fold into Linear(2,1)
            params[102] = (float)((double)Wf[0] * a0);
            params[103] = (float)((double)Wf[1] * a1);
            params[104] = (float)((double)bf[0] + (double)Wf[0] * c0 + (double)Wf[1] * c1);
        }
    }
}

__global__ void init_params(const float* __restrict__ W,
                            const float* __restrict__ bl,
                            float* __restrict__ params)
{
    if (threadIdx.x == 0 && blockIdx.x == 0) {
        params[0] = W[0]; params[1] = W[1]; params[2] = W[2]; params[3] = W[3];
        params[4] = bl[0]; params[5] = bl[1];
    }
}

extern "C" void kernel_launch(void* const* d_in, const int* in_sizes, int n_in,
                              void* d_out, int out_size, void* d_ws, size_t ws_size,
                              hipStream_t stream)
{
    const float* x  = (const float*)d_in[0];
    const float* W  = (const float*)d_in[1];
    const float* bl = (const float*)d_in[2];
    const float* gm = (const float*)d_in[3];
    const float* bt = (const float*)d_in[4];
    const float* sc = (const float*)d_in[5];
    const float* sh = (const float*)d_in[6];
    const float* Wf = (const float*)d_in[7];
    const float* bf = (const float*)d_in[8];
    float* out = (float*)d_out;

    const int nrows = in_sizes[0] / 2;          // [N,2] -> N
    const double invN = 1.0 / (double)nrows;

    // ws layout: [0,1024)   : folded params (105 floats used)
    //            [1024, ..) : per-block partials, NBLK*4 doubles (64 KB)
    float*  params   = (float*)d_ws;
    double* partials = (double*)((char*)d_ws + 1024);

    init_params<<<1, 64, 0, stream>>>(W, bl, params);

#define RUN_PASS(K)                                                              \
    chain_pass<(K) + 1, false><<<NBLK, NTHR, 0, stream>>>(                       \
        x, params, partials, nullptr, nrows);                                    \
    fold_kernel<<<1, NTHR, 0, stream>>>(partials, NBLK, W, bl, gm, bt, sc, sh,   \
                                        Wf, bf, params, (K), invN);

    RUN_PASS(0)  RUN_PASS(1)  RUN_PASS(2)  RUN_PASS(3)  RUN_PASS(4)
    RUN_PASS(5)  RUN_PASS(6)  RUN_PASS(7)  RUN_PASS(8)  RUN_PASS(9)
    RUN_PASS(10) RUN_PASS(11) RUN_PASS(12) RUN_PASS(13) RUN_PASS(14)
    RUN_PASS(15) RUN_PASS(16)
#undef RUN_PASS

    chain_pass<17, true><<<NBLK, NTHR, 0, stream>>>(x, params, nullptr, out, nrows);
}